// HeteroGNN_65695819759749
// MI455X (gfx1250) — compile-verified
//
#include <hip/hip_runtime.h>

// HeteroGNN on MI455X (gfx1250), fp32 WMMA path.
//
// Algebraic rewrite: segment_sum of per-edge Linear(concat[x_src,x_dst]) ==
//   (scatter_sum x_src) @ W_top + deg ⊙ (x_dst @ W_bot + b)
// so all edge-wise GEMM work becomes one feature scatter (L2-resident atomics)
// plus dense [N,128]x[128,128] GEMMs done with v_wmma_f32_16x16x4_f32.

typedef __attribute__((ext_vector_type(2))) float v2f;
typedef __attribute__((ext_vector_type(8))) float v8f;

#define H 128
#define LDSPAD 132          // LDS row pitch (floats) -> conflict-free b64 reads
#define TPB 8               // 16-row tiles per workgroup (8 waves)

// ---------------------------------------------------------------- utilities
__global__ __launch_bounds__(256) void zero_kernel(float* p, long n) {
  long i = (long)blockIdx.x * blockDim.x + threadIdx.x;
  if (i < n) p[i] = 0.0f;
}

__global__ __launch_bounds__(256) void pad_kernel(const float* __restrict__ in,
                                                  float* __restrict__ out,
                                                  int N, int F) {
  long t = (long)blockIdx.x * blockDim.x + threadIdx.x;
  if (t >= (long)N * H) return;
  int i = (int)(t >> 7), h = (int)(t & 127);
  out[t] = (h < F) ? in[(long)i * F + h] : 0.0f;
}

__global__ __launch_bounds__(256) void deg_kernel(const int* __restrict__ edge,
                                                  int E, float* __restrict__ deg) {
  int e = blockIdx.x * blockDim.x + threadIdx.x;
  if (e < E) atomicAdd(&deg[edge[E + e]], 1.0f);
}

// scatter_sum of 128-float feature rows: S[dst[e],:] += X[src[e],:]
// one wave per edge (32 lanes x float4 = 512B coalesced)
__global__ __launch_bounds__(256) void scatter_kernel(const float* __restrict__ X,
                                                      const int* __restrict__ edge,
                                                      int E, float* __restrict__ S) {
  long idx = (long)blockIdx.x * blockDim.x + threadIdx.x;
  if (idx >= (long)E * 32) return;
  int e = (int)(idx >> 5);
  int c = (int)(idx & 31) << 2;
  int s = edge[e], d = edge[E + e];
  const float4 v = *(const float4*)(X + (long)s * H + c);
  float* o = S + (long)d * H + c;
  atomicAdd(o + 0, v.x); atomicAdd(o + 1, v.y);
  atomicAdd(o + 2, v.z); atomicAdd(o + 3, v.w);
}

// ------------------------------------------------------------- dual GEMM
// Y[M,128] = act( A @ WA  +  rowscale ⊙ (B @ WB + bias) )
//   rowscale = deg[row] if HAS_DEG else 1.0 ; act = relu if RELU.
// WA/WB are [128,128] row-major. M must be a multiple of 16.
template <bool HAS_DEG, bool RELU>
__global__ __launch_bounds__(256) void gemm_dual_kernel(
    const float* __restrict__ A, const float* __restrict__ B,
    const float* __restrict__ WA, const float* __restrict__ WB,
    const float* __restrict__ bias, const float* __restrict__ deg,
    float* __restrict__ Y, int M) {
  // Stage both weight matrices transposed into LDS: lws[n*LDSPAD + k] = W[k][n]
  __shared__ float lwa[H * LDSPAD];
  __shared__ float lwb[H * LDSPAD];
  for (int idx = threadIdx.x; idx < H * H; idx += 256) {
    int k = idx >> 7, n = idx & 127;           // global read coalesced in n
    lwa[n * LDSPAD + k] = WA[idx];
    lwb[n * LDSPAD + k] = WB[idx];
  }
  __syncthreads();

  const int wave = threadIdx.x >> 5;
  const int lane = threadIdx.x & 31;
  const int l15  = lane & 15;
  const int kh2  = (lane >> 4) << 1;           // 0 for lanes 0-15, 2 for 16-31
  const int ntiles = M >> 4;

  for (int t = blockIdx.x * TPB + wave; t < ntiles; t += gridDim.x * TPB) {
    const int m0 = t << 4;
    v8f cA[8], cB[8];
#pragma unroll
    for (int n = 0; n < 8; ++n) {
      cA[n] = (v8f){0, 0, 0, 0, 0, 0, 0, 0};
      cB[n] = (v8f){0, 0, 0, 0, 0, 0, 0, 0};
    }

    // A-fragment: lane<16 holds K=k0,k0+1 of row m0+l15; lane>=16 holds K=k0+2,k0+3
    const float* Ap = A + (long)(m0 + l15) * H + kh2;
    const float* Bp = B + (long)(m0 + l15) * H + kh2;
    const float* wabase = &lwa[l15 * LDSPAD + kh2];
    const float* wbbase = &lwb[l15 * LDSPAD + kh2];

#pragma unroll 1
    for (int kk = 0; kk < 32; ++kk) {
      const int k0 = kk << 2;
      v2f a = *(const v2f*)(Ap + k0);
      v2f b = *(const v2f*)(Bp + k0);
#pragma unroll
      for (int n = 0; n < 8; ++n) {
        v2f wa = *(const v2f*)(wabase + n * 16 * LDSPAD + k0);
        v2f wb = *(const v2f*)(wbbase + n * 16 * LDSPAD + k0);
        cA[n] = __builtin_amdgcn_wmma_f32_16x16x4_f32(false, a, false, wa,
                                                      (short)0, cA[n], false, false);
        cB[n] = __builtin_amdgcn_wmma_f32_16x16x4_f32(false, b, false, wb,
                                                      (short)0, cB[n], false, false);
      }
    }

    // Epilogue. C layout: VGPR r = row m0+r (lanes 0-15) / m0+r+8 (lanes 16-31)
    const int rbase = m0 + ((lane >> 4) << 3);   // 8-aligned
    float degv[8];
    if (HAS_DEG) {
      const float4 d0 = *(const float4*)(deg + rbase);
      const float4 d1 = *(const float4*)(deg + rbase + 4);
      degv[0] = d0.x; degv[1] = d0.y; degv[2] = d0.z; degv[3] = d0.w;
      degv[4] = d1.x; degv[5] = d1.y; degv[6] = d1.z; degv[7] = d1.w;
    } else {
#pragma unroll
      for (int r = 0; r < 8; ++r) degv[r] = 1.0f;
    }
#pragma unroll
    for (int n = 0; n < 8; ++n) {
      const float bv = bias[n * 16 + l15];
#pragma unroll
      for (int r = 0; r < 8; ++r) {
        float acc = HAS_DEG ? (cA[n][r] + degv[r] * (cB[n][r] + bv))
                            : (cA[n][r] + cB[n][r] + bv);
        if (RELU) acc = fmaxf(acc, 0.0f);
        Y[(long)(rbase + r) * H + n * 16 + l15] = acc;
      }
    }
  }
}

// ---------------------------------------------------------------- pooling
__global__ __launch_bounds__(256) void pool_sum_kernel(const float* __restrict__ xl,
                                                       const int* __restrict__ batch,
                                                       int NL, float* __restrict__ sums) {
  long idx = (long)blockIdx.x * blockDim.x + threadIdx.x;
  if (idx >= (long)NL * 32) return;
  int link = (int)(idx >> 5);
  int c = (int)(idx & 31) << 2;
  int b = batch[link];
  const float4 v = *(const float4*)(xl + (long)link * H + c);
  float* o = sums + (long)b * H + c;
  atomicAdd(o + 0, v.x); atomicAdd(o + 1, v.y);
  atomicAdd(o + 2, v.z); atomicAdd(o + 3, v.w);
}

__global__ __launch_bounds__(256) void pool_cnt_kernel(const int* __restrict__ batch,
                                                       int NL, float* __restrict__ cnt) {
  int i = blockIdx.x * blockDim.x + threadIdx.x;
  if (i < NL) atomicAdd(&cnt[batch[i]], 1.0f);
}

__global__ __launch_bounds__(256) void finalize_kernel(const float* __restrict__ sums,
                                                       const float* __restrict__ cnt,
                                                       float* __restrict__ out, int G) {
  int t = blockIdx.x * blockDim.x + threadIdx.x;
  if (t >= G * H) return;
  int g = t >> 7;
  out[t] = sums[t] / fmaxf(cnt[g], 1.0f);
}

// ---------------------------------------------------------------- driver
extern "C" void kernel_launch(void* const* d_in, const int* in_sizes, int n_in,
                              void* d_out, int out_size, void* d_ws, size_t ws_size,
                              hipStream_t stream) {
  const float* x_node = (const float*)d_in[0];
  const float* x_link = (const float*)d_in[1];
  const float* Wm_nl  = (const float*)d_in[2];
  const float* bm_nl  = (const float*)d_in[3];
  const float* Wu_nl  = (const float*)d_in[4];
  const float* bu_nl  = (const float*)d_in[5];
  const float* Wm_ln  = (const float*)d_in[6];
  const float* bm_ln  = (const float*)d_in[7];
  const float* Wu_ln  = (const float*)d_in[8];
  const float* bu_ln  = (const float*)d_in[9];
  const int*   nl     = (const int*)d_in[10];
  const int*   ln     = (const int*)d_in[11];
  const int*   batch  = (const int*)d_in[12];

  const int NL = in_sizes[12];            // 50000 links
  const int NN = 50000;                   // problem constant
  const int F_NODE = in_sizes[0] / NN;
  const int F_LINK = in_sizes[1] / NL;
  const int E  = in_sizes[10] / 2;
  const int Lnum = in_sizes[3] / H;       // bm_nl = [L, H]
  const int G = out_size / H;

  // workspace layout
  const size_t MH = (size_t)((NN > NL) ? NN : NL) * H;
  float* ws   = (float*)d_ws;
  float* xn   = ws;
  float* xl   = xn  + MH;
  float* xn2  = xl  + MH;
  float* xl2  = xn2 + MH;
  float* Sl   = xl2 + MH;
  float* Sn   = Sl  + MH;
  float* AGG  = Sn  + MH;
  float* dgl  = AGG + MH;
  float* dgn  = dgl + NL;
  float* sums = dgn + NN;
  float* cnt  = sums + (size_t)G * H;
  (void)n_in; (void)ws_size;

  auto nb = [](long n, int b) { return (int)((n + b - 1) / b); };

  // degrees (constant across layers)
  zero_kernel<<<nb(NL, 256), 256, 0, stream>>>(dgl, NL);
  zero_kernel<<<nb(NN, 256), 256, 0, stream>>>(dgn, NN);
  deg_kernel<<<nb(E, 256), 256, 0, stream>>>(nl, E, dgl);
  deg_kernel<<<nb(E, 256), 256, 0, stream>>>(ln, E, dgn);

  // pad raw features to H
  pad_kernel<<<nb((long)NN * H, 256), 256, 0, stream>>>(x_node, xn, NN, F_NODE);
  pad_kernel<<<nb((long)NL * H, 256), 256, 0, stream>>>(x_link, xl, NL, F_LINK);

  for (int l = 0; l < Lnum; ++l) {
    float* cxn = (l & 1) ? xn2 : xn;
    float* cxl = (l & 1) ? xl2 : xl;
    float* oxn = (l & 1) ? xn  : xn2;
    float* oxl = (l & 1) ? xl  : xl2;

    const float* WmNLt = Wm_nl + (size_t)l * 2 * H * H;   // top rows (src = node)
    const float* WmNLb = WmNLt + H * H;                   // bottom rows (dst = link)
    const float* WuNLt = Wu_nl + (size_t)l * 2 * H * H;
    const float* WuNLb = WuNLt + H * H;
    const float* WmLNt = Wm_ln + (size_t)l * 2 * H * H;
    const float* WmLNb = WmLNt + H * H;
    const float* WuLNt = Wu_ln + (size_t)l * 2 * H * H;
    const float* WuLNb = WuLNt + H * H;

    zero_kernel<<<nb((long)NL * H, 256), 256, 0, stream>>>(Sl, (long)NL * H);
    zero_kernel<<<nb((long)NN * H, 256), 256, 0, stream>>>(Sn, (long)NN * H);
    scatter_kernel<<<nb((long)E * 32, 256), 256, 0, stream>>>(cxn, nl, E, Sl);
    scatter_kernel<<<nb((long)E * 32, 256), 256, 0, stream>>>(cxl, ln, E, Sn);

    const int gl = nb(NL / 16, TPB);
    const int gn = nb(NN / 16, TPB);
    // link update: agg = Sl@Wtop + deg⊙(xl@Wbot + bm); xl' = relu(agg@Wut + xl@Wub + bu)
    gemm_dual_kernel<true, false><<<gl, 256, 0, stream>>>(
        Sl, cxl, WmNLt, WmNLb, bm_nl + (size_t)l * H, dgl, AGG, NL);
    gemm_dual_kernel<false, true><<<gl, 256, 0, stream>>>(
        AGG, cxl, WuNLt, WuNLb, bu_nl + (size_t)l * H, nullptr, oxl, NL);
    // node update
    gemm_dual_kernel<true, false><<<gn, 256, 0, stream>>>(
        Sn, cxn, WmLNt, WmLNb, bm_ln + (size_t)l * H, dgn, AGG, NN);
    gemm_dual_kernel<false, true><<<gn, 256, 0, stream>>>(
        AGG, cxn, WuLNt, WuLNb, bu_ln + (size_t)l * H, nullptr, oxn, NN);
  }

  float* fxl = (Lnum & 1) ? xl2 : xl;
  zero_kernel<<<nb((long)G * H, 256), 256, 0, stream>>>(sums, (long)G * H);
  zero_kernel<<<nb(G, 256), 256, 0, stream>>>(cnt, G);
  pool_sum_kernel<<<nb((long)NL * 32, 256), 256, 0, stream>>>(fxl, batch, NL, sums);
  pool_cnt_kernel<<<nb(NL, 256), 256, 0, stream>>>(batch, NL, cnt);
  finalize_kernel<<<nb((long)G * H, 256), 256, 0, stream>>>(sums, cnt, (float*)d_out, G);
}